// SpatialGate_45896020525452
// MI455X (gfx1250) — compile-verified
//
#include <hip/hip_runtime.h>
#include <math.h>

typedef float v2f __attribute__((ext_vector_type(2)));
typedef float v4f __attribute__((ext_vector_type(4)));
typedef float v8f __attribute__((ext_vector_type(8)));
typedef int   v4i __attribute__((ext_vector_type(4)));

#define HW   65536   // 256*256
#define NCH  64
#define NB   16
// per-batch accumulator layout (8 floats):
// [0]=sum_max [1]=sum_mean [2]=sum_min [3]=cnt [4]=ss_max [5]=ss_mean [6]=ss_min [7]=pad

__global__ void zero_accum_kernel(float* __restrict__ a) {
    a[threadIdx.x] = 0.0f;  // 128 = 16*8 floats
}

// ---------------------------------------------------------------------------
// Pass 1: channel pool (max / mean-via-WMMA / min) + fused masked block sums.
// grid = (64, 16), block = 256. Each thread owns 4 contiguous pixels (b128).
// x is a 268 MB one-shot stream -> non-temporal loads so it does not evict
// the 12 MB pooled tensor (stored RT, stays dirty in L2 for passes 2/3).
// ---------------------------------------------------------------------------
__global__ __launch_bounds__(256) void cp_pool_kernel(
    const float* __restrict__ x, const int* __restrict__ mask,
    float* __restrict__ out, float* __restrict__ accum)
{
    const int tid  = threadIdx.x;
    const int lane = tid & 31;
    const int b    = blockIdx.y;
    const int pix0 = blockIdx.x * 1024 + tid * 4;

    // A matrix (16x4 f32): rows 0-7 = [1,1,0,0]/64, rows 8-15 = [0,0,1,1]/64.
    // In the 16x4 f32 VGPR layout this makes D[.,n] land in the lane that
    // supplied B[.,n], i.e. a pure per-lane 2-channel accumulate per WMMA.
    const float aval =
        (((lane >> 3) & 1) == ((lane >> 4) & 1)) ? (1.0f / 64.0f) : 0.0f;
    const v2f A = { aval, aval };

    const float* xb = x + (size_t)b * NCH * HW + pix0;

    v4f vmax = { -3.4e38f, -3.4e38f, -3.4e38f, -3.4e38f };
    v4f vmin = {  3.4e38f,  3.4e38f,  3.4e38f,  3.4e38f };
    v8f c0 = {}, c1 = {}, c2 = {}, c3 = {};

    for (int c = 0; c < NCH; c += 2) {
        v4f xa = __builtin_nontemporal_load((const v4f*)(xb + (size_t)c       * HW));
        v4f xc = __builtin_nontemporal_load((const v4f*)(xb + (size_t)(c + 1) * HW));

        vmax = __builtin_elementwise_max(vmax, __builtin_elementwise_max(xa, xc));
        vmin = __builtin_elementwise_min(vmin, __builtin_elementwise_min(xa, xc));

        v2f b0 = { xa.x, xc.x };
        v2f b1 = { xa.y, xc.y };
        v2f b2 = { xa.z, xc.z };
        v2f b3 = { xa.w, xc.w };
        c0 = __builtin_amdgcn_wmma_f32_16x16x4_f32(false, A, false, b0, (short)0, c0, false, false);
        c1 = __builtin_amdgcn_wmma_f32_16x16x4_f32(false, A, false, b1, (short)0, c1, false, false);
        c2 = __builtin_amdgcn_wmma_f32_16x16x4_f32(false, A, false, b2, (short)0, c2, false, false);
        c3 = __builtin_amdgcn_wmma_f32_16x16x4_f32(false, A, false, b3, (short)0, c3, false, false);
    }

    // element 0 of each accumulator holds this lane's channel mean
    v4f vmean = { c0[0], c1[0], c2[0], c3[0] };

    float* ob = out + (size_t)b * 3 * HW + pix0;
    *(v4f*)(ob)          = vmax;   // stat 0: max   (RT -> hot in L2 for pass 2/3)
    *(v4f*)(ob + HW)     = vmean;  // stat 1: mean
    *(v4f*)(ob + 2 * HW) = vmin;   // stat 2: min

    // fused masked partial sums for ChannelNorm mean
    v4i m4 = *(const v4i*)(mask + (size_t)b * HW + pix0);
    v4f mf = { m4.x == 1 ? 1.0f : 0.0f, m4.y == 1 ? 1.0f : 0.0f,
               m4.z == 1 ? 1.0f : 0.0f, m4.w == 1 ? 1.0f : 0.0f };
    v4f pmx = vmax * mf, pme = vmean * mf, pmn = vmin * mf;
    v4f part = { pmx.x + pmx.y + pmx.z + pmx.w,
                 pme.x + pme.y + pme.z + pme.w,
                 pmn.x + pmn.y + pmn.z + pmn.w,
                 mf.x + mf.y + mf.z + mf.w };

    __shared__ v4f red[256];
    red[tid] = part;
    __syncthreads();
    for (int off = 128; off > 0; off >>= 1) {
        if (tid < off) red[tid] += red[tid + off];
        __syncthreads();
    }
    if (tid == 0) {
        v4f r = red[0];
        atomicAdd(&accum[b * 8 + 0], r.x);
        atomicAdd(&accum[b * 8 + 1], r.y);
        atomicAdd(&accum[b * 8 + 2], r.z);
        atomicAdd(&accum[b * 8 + 3], r.w);
    }
}

// ---------------------------------------------------------------------------
// Pass 2: masked sum of squared deviations per (b, stat). L2-resident.
// ---------------------------------------------------------------------------
__global__ __launch_bounds__(256) void cn_var_kernel(
    const float* __restrict__ out, const int* __restrict__ mask,
    float* __restrict__ accum)
{
    const int tid  = threadIdx.x;
    const int b    = blockIdx.y;
    const int pix0 = blockIdx.x * 1024 + tid * 4;

    const float cnt = accum[b * 8 + 3];
    const float inv_cnt = 1.0f / cnt;
    float mu0 = accum[b * 8 + 0] * inv_cnt;
    float mu1 = accum[b * 8 + 1] * inv_cnt;
    float mu2 = accum[b * 8 + 2] * inv_cnt;

    v4i m4 = *(const v4i*)(mask + (size_t)b * HW + pix0);
    v4f mf = { m4.x == 1 ? 1.0f : 0.0f, m4.y == 1 ? 1.0f : 0.0f,
               m4.z == 1 ? 1.0f : 0.0f, m4.w == 1 ? 1.0f : 0.0f };

    const float* ob = out + (size_t)b * 3 * HW + pix0;
    v4f v0 = *(const v4f*)(ob);
    v4f v1 = *(const v4f*)(ob + HW);
    v4f v2 = *(const v4f*)(ob + 2 * HW);
    v4f d0 = (v0 - mu0); d0 = d0 * d0 * mf;
    v4f d1 = (v1 - mu1); d1 = d1 * d1 * mf;
    v4f d2 = (v2 - mu2); d2 = d2 * d2 * mf;

    __shared__ v4f red[256];
    red[tid] = (v4f){ d0.x + d0.y + d0.z + d0.w,
                      d1.x + d1.y + d1.z + d1.w,
                      d2.x + d2.y + d2.z + d2.w, 0.0f };
    __syncthreads();
    for (int off = 128; off > 0; off >>= 1) {
        if (tid < off) red[tid] += red[tid + off];
        __syncthreads();
    }
    if (tid == 0) {
        v4f r = red[0];
        atomicAdd(&accum[b * 8 + 4], r.x);
        atomicAdd(&accum[b * 8 + 5], r.y);
        atomicAdd(&accum[b * 8 + 6], r.z);
    }
}

// ---------------------------------------------------------------------------
// Pass 3: in-place normalize + mask. L2-resident read; final store is NT
// (never re-read).
// ---------------------------------------------------------------------------
__global__ __launch_bounds__(256) void cn_norm_kernel(
    float* __restrict__ out, const int* __restrict__ mask,
    const float* __restrict__ accum)
{
    const int tid  = threadIdx.x;
    const int b    = blockIdx.y;
    const int pix0 = blockIdx.x * 1024 + tid * 4;

    const float cnt = accum[b * 8 + 3];
    const float inv_cnt = 1.0f / cnt;
    const float inv_n1  = 1.0f / (cnt - 1.0f);

    v4i m4 = *(const v4i*)(mask + (size_t)b * HW + pix0);
    v4f mf = { m4.x == 1 ? 1.0f : 0.0f, m4.y == 1 ? 1.0f : 0.0f,
               m4.z == 1 ? 1.0f : 0.0f, m4.w == 1 ? 1.0f : 0.0f };

    float* ob = out + (size_t)b * 3 * HW + pix0;
    #pragma unroll
    for (int s = 0; s < 3; ++s) {
        float mu   = accum[b * 8 + s] * inv_cnt;
        float rstd = 1.0f / sqrtf(accum[b * 8 + 4 + s] * inv_n1);
        v4f v = *(const v4f*)(ob + (size_t)s * HW);
        v = (v - mu) * rstd * mf;
        __builtin_nontemporal_store(v, (v4f*)(ob + (size_t)s * HW));
    }
}

extern "C" void kernel_launch(void* const* d_in, const int* in_sizes, int n_in,
                              void* d_out, int out_size, void* d_ws, size_t ws_size,
                              hipStream_t stream) {
    const float* x    = (const float*)d_in[0];
    const int*   mask = (const int*)d_in[1];
    float*       out  = (float*)d_out;
    float*       acc  = (float*)d_ws;   // 16 batches * 8 floats = 512 B

    zero_accum_kernel<<<1, NB * 8, 0, stream>>>(acc);

    dim3 grid(HW / 1024, NB);           // (64, 16)
    cp_pool_kernel<<<grid, 256, 0, stream>>>(x, mask, out, acc);
    cn_var_kernel <<<grid, 256, 0, stream>>>(out, mask, acc);
    cn_norm_kernel<<<grid, 256, 0, stream>>>(out, mask, acc);
}